// ToyNet_20426864459907
// MI455X (gfx1250) — compile-verified
//
#include <hip/hip_runtime.h>
#include <math.h>

typedef float v2f __attribute__((ext_vector_type(2)));
typedef float v8f __attribute__((ext_vector_type(8)));
typedef int   i32x4 __attribute__((vector_size(16)));

#define F_IN 512
#define HID  16
#define NCLS 40
#define KCH  32                  // K-chunk staged per async round
#define XPAD 36                  // padded row stride (floats): 144B -> 16B aligned,
                                 // banks 4*(9m mod 16): all 16 rows distinct

// ---- CDNA5 async global->LDS (ASYNCcnt) with portable fallback ------------
#if __has_builtin(__builtin_amdgcn_global_load_async_to_lds_b128) && \
    __has_builtin(__builtin_amdgcn_s_wait_asynccnt)
#define ASYNC_COPY_B128(g, l)                                                  \
    __builtin_amdgcn_global_load_async_to_lds_b128(                            \
        (__attribute__((address_space(1))) i32x4*)(g),                         \
        (__attribute__((address_space(3))) i32x4*)(l), 0, 0)
#define WAIT_ASYNC(n) __builtin_amdgcn_s_wait_asynccnt(n)
#else
#define ASYNC_COPY_B128(g, l)                                                  \
    do { *(float4*)(l) = *(const float4*)(g); } while (0)
#define WAIT_ASYNC(n) do { } while (0)
#endif

// ---------------------------------------------------------------------------
// Fused GEMM1 + GEMM2:
//   h1 = relu(x @ W1^T + b1)   (regs/LDS only)
//   hg = h1 @ Wg^T             (global)
// One wave32 per 16-node tile; V_WMMA_F32_16X16X4_F32 accumulation.
// x is staged per-wave into LDS in 16xKCH chunks via coalesced async b128
// (each b128 instr: 32 lanes x 16B = 4 full rows of the chunk, every
// cacheline fetched exactly once), double-buffered against the WMMA loop.
// ---------------------------------------------------------------------------
__global__ void __launch_bounds__(256)
k_gemm_fused(const float* __restrict__ x, const float* __restrict__ W1,
             const float* __restrict__ b1, const float* __restrict__ Wg,
             float* __restrict__ hg, int N)
{
    __shared__ __align__(16) float xs[8][2][16 * XPAD];  // 36.0 KB
    __shared__ float tr[8][16 * 16];                     //  8.0 KB
    const int lane  = threadIdx.x & 31;
    const int wave  = threadIdx.x >> 5;
    const int row0  = (blockIdx.x * 8 + wave) * 16;
    if (row0 + 16 > N) return;                // uniform per wave (N%16==0)

    const int m     = lane & 15;              // A row / B col
    const int khalf = lane >> 4;              // 0: K={0,1}, 1: K={2,3}

    // coalesced staging pattern: instr i covers chunk rows 4i..4i+3
    const int srow = lane >> 3;               // 0..3
    const int scol = (lane & 7) * 4;          // float col within chunk
    const float* gs0 = x + (size_t)(row0 + srow) * F_IN + scol;
    float* ls0 = &xs[wave][0][srow * XPAD + scol];
    float* ls1 = &xs[wave][1][srow * XPAD + scol];

    // prologue: chunk 0
    #pragma unroll
    for (int i = 0; i < 4; ++i)
        ASYNC_COPY_B128(gs0 + (size_t)(4 * i) * F_IN, ls0 + 4 * i * XPAD);

    const float* __restrict__ bw = W1 + (size_t)m * F_IN + 2 * khalf;

    v8f c = {};
    #pragma unroll 2
    for (int ch = 0; ch < F_IN / KCH; ++ch) {
        float* lbuf = (ch & 1) ? &xs[wave][1][0] : &xs[wave][0][0];
        if (ch + 1 < F_IN / KCH) {
            const float* gs = gs0 + (ch + 1) * KCH;
            float* ld = (ch & 1) ? ls0 : ls1;
            #pragma unroll
            for (int i = 0; i < 4; ++i)
                ASYNC_COPY_B128(gs + (size_t)(4 * i) * F_IN, ld + 4 * i * XPAD);
            WAIT_ASYNC(4);          // in-order: current chunk's 4 have landed
        } else {
            WAIT_ASYNC(0);
        }
        const float* lw = lbuf + m * XPAD + 2 * khalf;
        const float* bk = bw + ch * KCH;
        #pragma unroll
        for (int k = 0; k < KCH; k += 4) {
            v2f a = *(const v2f*)(lw + k);
            v2f b; b.x = bk[k]; b.y = bk[k + 1];
            c = __builtin_amdgcn_wmma_f32_16x16x4_f32(false, a, false, b,
                                                      (short)0, c, false, false);
        }
    }

    // bias + relu; park tile in wave-private LDS: C-layout -> A-layout
    const float bias = b1[m];
    float* lt = tr[wave];
    #pragma unroll
    for (int r = 0; r < 8; ++r) {
        float v = c[r] + bias;
        lt[(r + khalf * 8) * 16 + m] = v > 0.f ? v : 0.f;
    }

    // GEMM2: 16x16 @ 16x16, K=16 -> 4 WMMA steps
    v8f c2 = {};
    #pragma unroll
    for (int k0 = 0; k0 < HID; k0 += 4) {
        v2f a, b;
        a.x = lt[m * 16 + k0 + 2 * khalf];
        a.y = lt[m * 16 + k0 + 2 * khalf + 1];
        b.x = Wg[m * 16 + k0 + 2 * khalf];
        b.y = Wg[m * 16 + k0 + 2 * khalf + 1];
        c2 = __builtin_amdgcn_wmma_f32_16x16x4_f32(false, a, false, b,
                                                   (short)0, c2, false, false);
    }
    #pragma unroll
    for (int r = 0; r < 8; ++r)
        hg[(size_t)(row0 + r + khalf * 8) * 16 + m] = c2[r];
}

// deg starts at 1.0 (self-loop); agg zeroed
__global__ void k_init(float* __restrict__ deg, float* __restrict__ agg, int N)
{
    int i = blockIdx.x * blockDim.x + threadIdx.x;
    if (i < N) deg[i] = 1.0f;
    if (i < N * HID) agg[i] = 0.0f;
}

__global__ void k_deg(const int* __restrict__ dst, float* __restrict__ deg, int E)
{
    int e = blockIdx.x * blockDim.x + threadIdx.x;
    if (e < E) atomicAdd(&deg[dst[e]], 1.0f);
}

__global__ void k_dinv(const float* __restrict__ deg, float* __restrict__ dinv, int N)
{
    int i = blockIdx.x * blockDim.x + threadIdx.x;
    if (i < N) {
        float d = deg[i];
        dinv[i] = d > 0.f ? rsqrtf(d) : 0.f;
    }
}

// agg[dst] += hg[src] * dinv[src] * dinv[dst]
// lane-per-(edge,feature): every atomic instruction covers 2 contiguous 64B
// rows instead of 32 scattered lines; hg gather is contiguous per edge.
__global__ void k_agg(const int* __restrict__ src, const int* __restrict__ dst,
                      const float* __restrict__ dinv, const float* __restrict__ hg,
                      float* __restrict__ agg, long long E)
{
    long long t = (long long)blockIdx.x * blockDim.x + threadIdx.x;
    long long e = t >> 4;
    if (e >= E) return;
    int f = (int)t & 15;
    int s = src[e], d = dst[e];
    float nrm = dinv[s] * dinv[d];
    atomicAdd(&agg[(size_t)d * HID + f], hg[(size_t)s * HID + f] * nrm);
}

// self-loop term + bias + relu + 16->40 classifier + log_softmax
__global__ void k_final(const float* __restrict__ agg, const float* __restrict__ hg,
                        const float* __restrict__ dinv, const float* __restrict__ bg,
                        const float* __restrict__ W2, const float* __restrict__ b2,
                        float* __restrict__ out, int N)
{
    int i = blockIdx.x * blockDim.x + threadIdx.x;
    if (i >= N) return;
    float sw = dinv[i] * dinv[i];
    float h[HID];
    #pragma unroll
    for (int f = 0; f < HID; ++f) {
        float v = agg[(size_t)i * HID + f] + hg[(size_t)i * HID + f] * sw + bg[f];
        h[f] = v > 0.f ? v : 0.f;
    }
    float lg[NCLS];
    float mx = -3.402823466e+38f;
    #pragma unroll
    for (int c = 0; c < NCLS; ++c) {
        float s = b2[c];
        #pragma unroll
        for (int f = 0; f < HID; ++f) s += W2[c * HID + f] * h[f];
        lg[c] = s;
        mx = s > mx ? s : mx;
    }
    float sum = 0.f;
    #pragma unroll
    for (int c = 0; c < NCLS; ++c) sum += __expf(lg[c] - mx);
    float lse = mx + __logf(sum);
    #pragma unroll
    for (int c = 0; c < NCLS; ++c) out[(size_t)i * NCLS + c] = lg[c] - lse;
}

extern "C" void kernel_launch(void* const* d_in, const int* in_sizes, int n_in,
                              void* d_out, int out_size, void* d_ws, size_t ws_size,
                              hipStream_t stream)
{
    const float* x  = (const float*)d_in[0];
    const int*   ei = (const int*)  d_in[1];   // [2, E]
    const float* W1 = (const float*)d_in[2];
    const float* b1 = (const float*)d_in[3];
    const float* Wg = (const float*)d_in[4];
    const float* bg = (const float*)d_in[5];
    const float* W2 = (const float*)d_in[6];
    const float* b2 = (const float*)d_in[7];

    const int N = in_sizes[0] / F_IN;
    const int E = in_sizes[1] / 2;
    const int* src = ei;
    const int* dst = ei + E;

    float* ws   = (float*)d_ws;
    float* deg  = ws;                       // N
    float* dinv = ws + N;                   // N
    float* hg   = ws + 2 * (size_t)N;       // N*16
    float* agg  = hg + (size_t)N * HID;     // N*16
    float* out  = (float*)d_out;            // N*40

    const int T = 256;
    k_init <<<(N * HID + T - 1) / T, T, 0, stream>>>(deg, agg, N);
    k_deg  <<<(E + T - 1) / T,       T, 0, stream>>>(dst, deg, E);
    k_dinv <<<(N + T - 1) / T,       T, 0, stream>>>(deg, dinv, N);

    const int tiles = (N + 15) / 16;                 // one wave per tile
    k_gemm_fused<<<(tiles + 7) / 8, 256, 0, stream>>>(x, W1, b1, Wg, hg, N);

    long long work = (long long)E * HID;             // lane per (edge, feature)
    k_agg  <<<(unsigned)((work + T - 1) / T), T, 0, stream>>>(src, dst, dinv, hg, agg, E);
    k_final<<<(N + T - 1) / T, T, 0, stream>>>(agg, hg, dinv, bg, W2, b2, out, N);
}